// WeightedStateLoss5_46995532153320
// MI455X (gfx1250) — compile-verified
//
#include <hip/hip_runtime.h>

// Vector types
typedef __attribute__((ext_vector_type(4))) float v4f;
typedef __attribute__((ext_vector_type(2))) float v2f;
typedef __attribute__((ext_vector_type(8))) float v8f;

// Wave32 cross-lane reduction via V_WMMA_F32_16X16X4_F32.
// Each lane contributes two partials (A-matrix 16x4 f32 = 2 VGPRs/lane).
// B = all-ones 4x16  =>  D[m][n] = rowsum(A row m) for every n.
// C/D layout: lane l in [0,16): VGPR v = D[v][l]      = rowsum(v)
//             lane l in [16,32): VGPR v = D[v+8][l-16] = rowsum(v+8)
// Per-lane sum of the 8 D VGPRs => lanes 0-15 hold Sum(rows 0..7),
// lanes 16-31 hold Sum(rows 8..15). One xor-16 shuffle completes the total.
__device__ __forceinline__ float wave_reduce_wmma(float acc0, float acc1) {
    v2f a; a[0] = acc0; a[1] = acc1;
    v2f b; b[0] = 1.0f; b[1] = 1.0f;
    v8f c = {};
    c = __builtin_amdgcn_wmma_f32_16x16x4_f32(
        /*neg_a=*/false, a, /*neg_b=*/false, b,
        /*c_mod=*/(short)0, c, /*reuse_a=*/false, /*reuse_b=*/false);
    float s = ((c[0] + c[1]) + (c[2] + c[3])) + ((c[4] + c[5]) + (c[6] + c[7]));
    s += __shfl_xor(s, 16, 32);   // combine lane-group 0-15 with 16-31
    return s;                     // every lane now holds the wave total
}

// Stage 1: grid-stride streaming of pred/targ as float4 with non-temporal
// hint (read-once 210 MB; don't pollute L2), masked squared error, then
// WMMA wave reduce + LDS combine -> one partial per block.
__global__ void masked_sqerr_partial(const v4f* __restrict__ pred,
                                     const v4f* __restrict__ targ,
                                     float* __restrict__ partial,
                                     int n4) {
    const int tid    = blockIdx.x * blockDim.x + threadIdx.x;
    const int stride = gridDim.x * blockDim.x;

    float acc0 = 0.0f, acc1 = 0.0f;
    for (int i = tid; i < n4; i += stride) {
        v4f p = __builtin_nontemporal_load(&pred[i]);
        v4f t = __builtin_nontemporal_load(&targ[i]);
        float dx = p[0] - t[0];
        float dy = p[1] - t[1];
        float dz = p[2] - t[2];
        float dw = p[3] - t[3];
        acc0 += (t[0] != 0.0f ? dx * dx : 0.0f) + (t[1] != 0.0f ? dy * dy : 0.0f);
        acc1 += (t[2] != 0.0f ? dz * dz : 0.0f) + (t[3] != 0.0f ? dw * dw : 0.0f);
    }

    // All lanes reach here (uniform loop bound check) -> EXEC all ones for WMMA.
    float s = wave_reduce_wmma(acc0, acc1);

    __shared__ float lds[8];                  // 256 threads = 8 wave32s
    const int lane = threadIdx.x & 31;
    const int wid  = threadIdx.x >> 5;
    if (lane == 0) lds[wid] = s;
    __syncthreads();
    if (threadIdx.x == 0) {
        float tot = 0.0f;
        const int nw = (int)(blockDim.x >> 5);
        for (int w = 0; w < nw; ++w) tot += lds[w];   // fixed order: deterministic
        partial[blockIdx.x] = tot;
    }
}

// Stage 2: one block reduces all block partials in a fixed order, scales by
// 1/B, writes both output slots (loss, a0_loss).
__global__ void finalize_loss(const float* __restrict__ partial, int n,
                              float scale, float* __restrict__ out) {
    float acc0 = 0.0f, acc1 = 0.0f;
    for (int i = threadIdx.x; i < n; i += (int)blockDim.x)
        acc0 += partial[i];                  // fixed per-thread order

    float s = wave_reduce_wmma(acc0, acc1);

    __shared__ float lds[8];
    const int lane = threadIdx.x & 31;
    const int wid  = threadIdx.x >> 5;
    if (lane == 0) lds[wid] = s;
    __syncthreads();
    if (threadIdx.x == 0) {
        float tot = 0.0f;
        const int nw = (int)(blockDim.x >> 5);
        for (int w = 0; w < nw; ++w) tot += lds[w];
        tot *= scale;
        out[0] = tot;   // loss
        out[1] = tot;   // a0_loss
    }
}

extern "C" void kernel_launch(void* const* d_in, const int* in_sizes, int n_in,
                              void* d_out, int out_size, void* d_ws, size_t ws_size,
                              hipStream_t stream) {
    (void)n_in; (void)out_size; (void)ws_size;

    const float* pred = (const float*)d_in[0];
    const float* targ = (const float*)d_in[1];
    float*       out  = (float*)d_out;
    float*       part = (float*)d_ws;        // stage-1 partials (BLOCKS floats)

    const int n  = in_sizes[0];              // B*H*D = 26,214,400 (divisible by 4)
    const int n4 = n >> 2;

    // mean over batch: B = n / (H*D) = n / 6400
    const int   batch = n / 6400;
    const float scale = 1.0f / (float)batch;

    const int THREADS = 256;                 // 8 wave32s per block
    const int BLOCKS  = 2048;                // 16K waves in flight: saturate HBM

    masked_sqerr_partial<<<BLOCKS, THREADS, 0, stream>>>(
        (const v4f*)pred, (const v4f*)targ, part, n4);
    finalize_loss<<<1, THREADS, 0, stream>>>(part, BLOCKS, scale, out);
}